// LSTMClassifier_63367947485624
// MI455X (gfx1250) — compile-verified
//
#include <hip/hip_runtime.h>

// ---------------------------------------------------------------------------
// LSTM classifier, MI455X (gfx1250, wave32, WMMA).
//
//  * Input projection folded into a VOCAB x 32 LDS table -> only HBM traffic
//    is the 33.5 MB token stream (~1.5 us at 23.3 TB/s); kernel is bound by
//    the sequential recurrence, parallel over B=4096 chains (256 waves).
//  * Per wave (16 chains): gates^T[32x16] = W_hh[32x8] x h^T[8x16] via 4x
//    V_WMMA_F32_16X16X4_F32 per step. A(=W_hh) loop-invariant; C initialized
//    straight from LDS table (4x ds_load_b128).
//  * Paired-ownership elementwise stage: low lanes own hidden units
//    {0,1,4,5}, high lanes own {2,3,6,7}. Each v_tanh slot does work for
//    both halves (per-lane constant scale/bias turns the same instruction
//    into sig() on one half and tanh() on the other): 20 v_tanh + 8
//    v_permlanex16 per step, and the resulting per-lane h registers ARE the
//    next step's B-fragments (zero shuffle/select to close the recurrence).
//  * Tokens streamed as int4 with one-block-ahead prefetch (1 global load
//    per 4 steps).
// ---------------------------------------------------------------------------

typedef __attribute__((ext_vector_type(2))) float v2f;
typedef __attribute__((ext_vector_type(8))) float v8f;

#define EDIM 8
#define HDIM 8
#define G4   32          // 4*H gates
#define TROW 36          // padded LDS table row stride (dwords, 144B = 16B-aligned)

__device__ __forceinline__ float fast_exp2(float x) {
#if __has_builtin(__builtin_amdgcn_exp2f)
  return __builtin_amdgcn_exp2f(x);
#else
  return exp2f(x);
#endif
}
__device__ __forceinline__ float fast_rcp(float x) {
#if __has_builtin(__builtin_amdgcn_rcpf)
  return __builtin_amdgcn_rcpf(x);
#else
  return 1.0f / x;
#endif
}

// r = tanh(z): single V_TANH_F32 if available, else exp2/rcp core.
__device__ __forceinline__ float tanh_core(float z) {
#if __has_builtin(__builtin_amdgcn_tanhf)
  return __builtin_amdgcn_tanhf(z);
#elif __has_builtin(__builtin_amdgcn_tanh_f32)
  return __builtin_amdgcn_tanh_f32(z);
#else
  const float e = fast_exp2(-2.88539008177793f * z);
  return 2.0f * fast_rcp(1.0f + e) - 1.0f;
#endif
}
__device__ __forceinline__ float fsig(float x) {
  return 0.5f * tanh_core(0.5f * x) + 0.5f;
}

// Swap values between lane halves (lane ^ 16) as a pure VALU op.
__device__ __forceinline__ float xswap16(float v) {
#if __has_builtin(__builtin_amdgcn_permlanex16)
  const int r = __builtin_amdgcn_permlanex16(
      __float_as_int(v), __float_as_int(v), 0x76543210, 0xFEDCBA98, false, false);
  return __int_as_float(r);
#else
  return __shfl_xor(v, 16, 32);
#endif
}

__device__ __forceinline__ v8f wmma_f32_16x16x4(v2f a, v2f b, v8f c) {
  return __builtin_amdgcn_wmma_f32_16x16x4_f32(
      /*neg_a=*/false, a, /*neg_b=*/false, b,
      /*c_mod=*/(short)0, c, /*reuse_a=*/false, /*reuse_b=*/false);
}

__global__ __launch_bounds__(64) void lstm_cls_kernel(
    const int* __restrict__ x,       // [B, T]
    const float* __restrict__ emb,   // [VOCAB, E]
    const float* __restrict__ W_ih,  // [4H, E]
    const float* __restrict__ W_hh,  // [4H, H]
    const float* __restrict__ b_ih,  // [4H]
    const float* __restrict__ b_hh,  // [4H]
    const float* __restrict__ W_cls, // [1, H]
    const float* __restrict__ b_cls, // [1]
    float* __restrict__ out,         // [B, 1]
    int T, int vocab)
{
  extern __shared__ float tab[];     // [vocab][TROW]

  const int tid = threadIdx.x;

  // ---- Phase 1: fused input-projection table in LDS.
  // tab[v][g] = dot(emb[v], W_ih[g]) + b_ih[g] + b_hh[g]
  for (int k = tid; k < vocab * G4; k += blockDim.x) {
    const int v = k >> 5, g = k & 31;
    const float* er = emb + v * EDIM;
    const float* wr = W_ih + g * EDIM;
    float s = b_ih[g] + b_hh[g];
#pragma unroll
    for (int e = 0; e < EDIM; ++e) s += er[e] * wr[e];
    tab[v * TROW + g] = s;
  }
  __syncthreads();

  // ---- Phase 2: recurrence. One wave = 16 batch chains.
  const int lane = tid & 31;
  const int wave = tid >> 5;
  const int m    = lane & 15;        // matrix row / batch sub-index
  const bool lo  = lane < 16;
  const int b    = (blockIdx.x * (blockDim.x >> 5) + wave) * 16 + m;

  // Loop-invariant A fragments (W_hh). 16x4 f32 A layout:
  //   lanes 0-15 : V0=A[m][k0+0], V1=A[m][k0+1]
  //   lanes 16-31: V0=A[m][k0+2], V1=A[m][k0+3]
  const float* wif = W_hh + m * HDIM;          // tile_if: gate rows 0..15 (i,f)
  const float* wgo = W_hh + (16 + m) * HDIM;   // tile_go: gate rows 16..31 (g,o)
  v2f a_if_lo, a_if_hi, a_go_lo, a_go_hi;
  a_if_lo[0] = wif[lo ? 0 : 2]; a_if_lo[1] = wif[lo ? 1 : 3];
  a_if_hi[0] = wif[lo ? 4 : 6]; a_if_hi[1] = wif[lo ? 5 : 7];
  a_go_lo[0] = wgo[lo ? 0 : 2]; a_go_lo[1] = wgo[lo ? 1 : 3];
  a_go_hi[0] = wgo[lo ? 4 : 6]; a_go_hi[1] = wgo[lo ? 5 : 7];

  // C-tile table column bases: tile_if -> i(0..7)/f(8..15); tile_go -> g/o.
  const int cb_if = lo ? 0 : 8;
  const int cb_go = lo ? 16 : 24;

  // Per-lane activation constants: low lanes need tanh(g), high lanes sig(o);
  // both evaluate kA*tanh(kA*z)+cA on the same v_tanh instruction.
  const float kA = lo ? 1.0f : 0.5f;
  const float cA = lo ? 0.0f : 0.5f;

  // Paired state ownership: pair q -> hidden unit JL[q] on low lanes,
  // JH[q] on high lanes. hh[q] == (lo ? h[JL[q]] : h[JH[q]]) which is
  // exactly the B-fragment element layout for the next step's WMMA.
  float hh[4], cc[4];
#pragma unroll
  for (int q = 0; q < 4; ++q) { hh[q] = 0.0f; cc[q] = 0.0f; }

  auto step = [&](int idx) {
    const float* trow = tab + idx * TROW;
    const float4 q0 = *(const float4*)(trow + cb_if);
    const float4 q1 = *(const float4*)(trow + cb_if + 4);
    const float4 q2 = *(const float4*)(trow + cb_go);
    const float4 q3 = *(const float4*)(trow + cb_go + 4);
    v8f acc_if, acc_go;
    acc_if[0] = q0.x; acc_if[1] = q0.y; acc_if[2] = q0.z; acc_if[3] = q0.w;
    acc_if[4] = q1.x; acc_if[5] = q1.y; acc_if[6] = q1.z; acc_if[7] = q1.w;
    acc_go[0] = q2.x; acc_go[1] = q2.y; acc_go[2] = q2.z; acc_go[3] = q2.w;
    acc_go[4] = q3.x; acc_go[5] = q3.y; acc_go[6] = q3.z; acc_go[7] = q3.w;

    // B fragments = h^T : hh[] already holds them (no selects needed).
    v2f b_lo, b_hi;
    b_lo[0] = hh[0]; b_lo[1] = hh[1];
    b_hi[0] = hh[2]; b_hi[1] = hh[3];

    // gates^T = W_hh x h^T + table  (K=8 chained as 4+4)
    acc_if = wmma_f32_16x16x4(a_if_lo, b_lo, acc_if);
    acc_if = wmma_f32_16x16x4(a_if_hi, b_hi, acc_if);
    acc_go = wmma_f32_16x16x4(a_go_lo, b_lo, acc_go);
    acc_go = wmma_f32_16x16x4(a_go_hi, b_hi, acc_go);

    // Pair q: low lanes finish hidden unit jl, high lanes hidden unit jh.
    const int JL[4] = {0, 1, 4, 5};
    const int JH[4] = {2, 3, 6, 7};
#pragma unroll
    for (int q = 0; q < 4; ++q) {
      const int jl = JL[q], jh = JH[q];
      // Own-unit gates (consumed locally) and cross-unit gates (to exchange).
      const float A0 = lo ? acc_if[jl] : acc_if[jh];  // i_jl | f_jh
      const float A1 = lo ? acc_go[jl] : acc_go[jh];  // g_jl | o_jh
      const float A2 = lo ? acc_if[jh] : acc_if[jl];  // i_jh | f_jl
      const float A3 = lo ? acc_go[jh] : acc_go[jl];  // g_jh | o_jl
      const float t0 = fsig(A0);                      // sig(i_jl) | sig(f_jh)
      const float t1 = kA * tanh_core(kA * A1) + cA;  // tanh(g_jl)| sig(o_jh)
      const float t2 = fsig(A2);                      // sig(i_jh) | sig(f_jl)
      const float t3 = kA * tanh_core(kA * A3) + cA;  // tanh(g_jh)| sig(o_jl)
      const float mm = t2 * t3;                       // low: i*g for jh (payload)
      const float s0 = lo ? mm : t2;                  // send: ig_jh | sig(f_jl)
      const float s1 = lo ? mm : t3;                  // send:  (dup)| sig(o_jl)
      const float p0 = xswap16(s0);                   // recv: sig(f_jl) | ig_jh
      const float p1 = xswap16(s1);                   // recv: sig(o_jl) | ig_jh
      const float ig = lo ? t0 * t1 : p0;             // i*g of own unit
      const float ff = lo ? p0 : t0;                  // sig(f) of own unit
      const float oo = lo ? p1 : t1;                  // sig(o) of own unit
      cc[q] = ff * cc[q] + ig;
      hh[q] = oo * tanh_core(cc[q]);
    }
  };

  const int* xrow = x + (size_t)b * (size_t)T;

  if ((T & 3) == 0) {
    // Streamed int4 tokens with one-block-ahead prefetch.
    int4 tok = *(const int4*)xrow;
    for (int t0 = 0; t0 < T; t0 += 4) {
      const int nx = (t0 + 4 < T) ? (t0 + 4) : t0;
      const int4 tokn = *(const int4*)(xrow + nx);
      step(tok.x);
      step(tok.y);
      step(tok.z);
      step(tok.w);
      tok = tokn;
    }
  } else {
    for (int t = 0; t < T; ++t) step(xrow[t]);
  }

  // ---- Classifier head: out[b] = sigmoid(h . W_cls + b_cls)
  // Each half holds 4 of the 8 h values; per-lane weight selects + one
  // permlanex16 reduce to the full dot product on both halves.
  const float w0 = lo ? W_cls[0] : W_cls[2];
  const float w1 = lo ? W_cls[1] : W_cls[3];
  const float w2 = lo ? W_cls[4] : W_cls[6];
  const float w3 = lo ? W_cls[5] : W_cls[7];
  float part = hh[0] * w0 + hh[1] * w1 + hh[2] * w2 + hh[3] * w3;
  float s = part + xswap16(part) + b_cls[0];
  s = fsig(s);
  if (lo) out[b] = s;
}

extern "C" void kernel_launch(void* const* d_in, const int* in_sizes, int n_in,
                              void* d_out, int out_size, void* d_ws, size_t ws_size,
                              hipStream_t stream) {
  (void)n_in; (void)d_ws; (void)ws_size;
  const int*   x     = (const int*)  d_in[0];
  const float* emb   = (const float*)d_in[1];
  const float* W_ih  = (const float*)d_in[2];
  const float* W_hh  = (const float*)d_in[3];
  const float* b_ih  = (const float*)d_in[4];
  const float* b_hh  = (const float*)d_in[5];
  const float* W_cls = (const float*)d_in[6];
  const float* b_cls = (const float*)d_in[7];
  float* out = (float*)d_out;

  const int B     = out_size;              // 4096
  const int T     = in_sizes[0] / B;       // 2048
  const int vocab = in_sizes[1] / EDIM;    // 1000

  const int threads = 64;                  // 2 waves -> 32 chains per block
  const int blocks  = B / 32;              // 128
  const size_t smem = (size_t)vocab * TROW * sizeof(float);  // 144 KB

  lstm_cls_kernel<<<blocks, threads, smem, stream>>>(
      x, emb, W_ih, W_hh, b_ih, b_hh, W_cls, b_cls, out, T, vocab);
}